// QuantumEncoder_65481071408833
// MI455X (gfx1250) — compile-verified
//
#include <hip/hip_runtime.h>

typedef float v2f __attribute__((ext_vector_type(2)));
typedef float v4f __attribute__((ext_vector_type(4)));
typedef float v8f __attribute__((ext_vector_type(8)));

#define NQ 18
#define NCOMP 128  // 64 batches x {Re, Im} real components

// ---------------------------------------------------------------------------
// Build the 3 reps x 5 stage matrices.
// Rep r applies  C_chain * (x)RY(phi_r[w])  with phi_r = a[r] + b[r-1].
// Stage s is a dense real matrix over a qubit window:
//   s=0: dim16, qubits 0-3 :  C23 C12 C01 * KronRY(0..3)
//   s=1: dim32, qubits 3-7 :  C67 C56 C45 C34 * (I (x) KronRY(4..7))
//   s=2: dim32, qubits 7-11,  s=3: dim32, qubits 11-15 (analogous)
//   s=4: dim16, qubits 14-17: C(16,17) C(15,16) * (I (x) RY16 (x) RY17)
// CNOTs are applied as row permutations of the Kronecker matrix.
// ---------------------------------------------------------------------------
__global__ void qe_setup_matrices(const float* __restrict__ vp,
                                  float* __restrict__ Mtab) {
  __shared__ float phi[3][NQ];
  if (threadIdx.x < 3 * NQ) {
    int r = threadIdx.x / NQ, w = threadIdx.x % NQ;
    float v = vp[(r * NQ + w) * 2 + 0];              // a[r][w]
    if (r > 0) v += vp[((r - 1) * NQ + w) * 2 + 1];  // + b[r-1][w]
    phi[r][w] = v;
  }
  __syncthreads();
  const int dims[5]   = {16, 32, 32, 32, 16};
  const int q0s[5]    = {0, 3, 7, 11, 14};
  const int nbs[5]    = {4, 5, 5, 5, 4};
  const int rymask[5] = {0xF, 0x1E, 0x1E, 0x1E, 0xC};
  const int ncn[5]    = {3, 4, 4, 4, 2};
  const int cnc[5][4] = {{0,1,2,0},{0,1,2,3},{0,1,2,3},{0,1,2,3},{1,2,0,0}};
  const int offs[5]   = {0, 256, 1280, 2304, 3328};
  for (int r = 0; r < 3; ++r) {
    for (int s = 0; s < 5; ++s) {
      int dim = dims[s];
      float* M = Mtab + r * 3584 + offs[s];
      for (int e = threadIdx.x; e < dim * dim; e += blockDim.x) {
        int i = e / dim, j = e % dim;
        // row index under inverse of the CNOT sequence (chain order cnc[0..])
        int ii = i;
        for (int k = ncn[s] - 1; k >= 0; --k) {
          int cb = cnc[s][k];
          ii ^= ((ii >> cb) & 1) << (cb + 1);
        }
        float prod = 1.f;
        for (int t = 0; t < nbs[s]; ++t) {
          int it = (ii >> t) & 1, jt = (j >> t) & 1;
          if ((rymask[s] >> t) & 1) {
            float h = 0.5f * phi[r][q0s[s] + t];
            float cc = cosf(h), ss = sinf(h);
            prod *= (it == jt) ? cc : (it ? ss : -ss);  // RY = [[c,-s],[s,c]]
          } else if (it != jt) {
            prod = 0.f;
          }
        }
        M[e] = prod;
      }
    }
  }
}

// ---------------------------------------------------------------------------
// Initial state: |psi0> = (x)_w RX(x_w)|0> -> amplitude = (-i)^popcount * prod(mag)
// Re/Im parts stored as separate real vectors (components 2b, 2b+1).
// ---------------------------------------------------------------------------
__global__ void qe_init_state(const float* __restrict__ x,
                              float* __restrict__ state) {
  __shared__ float cs[NQ], sn[NQ];
  int b = blockIdx.y;
  if (threadIdx.x < NQ) {
    float h = 0.5f * x[b * 32 + threadIdx.x];
    cs[threadIdx.x] = cosf(h);
    sn[threadIdx.x] = sinf(h);
  }
  __syncthreads();
  float* re = state + ((long)(2 * b) << NQ);
  float* im = state + ((long)(2 * b + 1) << NQ);
  int base = blockIdx.x * 4096;
  for (int k = 0; k < 16; ++k) {
    int idx = base + (k << 8) + threadIdx.x;
    float mag = 1.f;
#pragma unroll
    for (int w = 0; w < NQ; ++w) mag *= ((idx >> w) & 1) ? sn[w] : cs[w];
    float rv = 0.f, iv = 0.f;
    switch (__popc(idx) & 3) {  // (-i)^p
      case 0: rv = mag; break;
      case 1: iv = -mag; break;
      case 2: rv = -mag; break;
      case 3: iv = mag; break;
    }
    re[idx] = rv;
    im[idx] = iv;
  }
}

// ---------------------------------------------------------------------------
// Apply one stage matrix (DIM=16 or 32) to all NCOMP real state vectors.
// In-place, block-diagonal: out[col, j] = sum_j' M[j,j'] in[col, j'] where the
// j bits live at positions [P, P+W).  One 16-column output tile per wave per
// iteration; K accumulated with v_wmma_f32_16x16x4_f32.  P is a template
// parameter so all element offsets fold into VMEM immediate offsets.
// ---------------------------------------------------------------------------
template <int DIM, int P>
__global__ void qe_stage(float* __restrict__ state,
                         const float* __restrict__ Mg) {
  constexpr int W = (DIM == 16) ? 4 : 5;
  constexpr int STRIDE = 1 << P;
  __shared__ float M[DIM * DIM];
  for (int e = threadIdx.x; e < DIM * DIM; e += blockDim.x) M[e] = Mg[e];
  __syncthreads();
  const int lane = threadIdx.x & 31;
  const int lo = lane & 15, hi = lane >> 4;
  const int wave = (blockIdx.x * blockDim.x + threadIdx.x) >> 5;
  const int nwaves = (gridDim.x * blockDim.x) >> 5;
  constexpr int colbits = NQ - W;
  constexpr long ntiles = ((long)NCOMP << colbits) >> 4;  // multiple of nwaves
  for (long t = wave; t < ntiles; t += nwaves) {
    long gm = (t << 4) + lo;  // this lane's column (N index = lo)
    long comp = gm >> colbits;
    long m = gm & ((1L << colbits) - 1);
    float* col =
        state + (comp << NQ) + ((m >> P) << (P + W)) + (m & (STRIDE - 1));
    {  // prefetch the next tile's column (uniform guard; keeps EXEC intact)
      long tn = t + nwaves;
      if (tn < ntiles) {
        long g2 = (tn << 4) + lo;
        long c2 = g2 >> colbits;
        long m2 = g2 & ((1L << colbits) - 1);
        __builtin_prefetch(
            state + (c2 << NQ) + ((m2 >> P) << (P + W)) + (m2 & (STRIDE - 1)),
            0, 1);
      }
    }
    if constexpr (DIM == 16) {
      v8f c = {0.f, 0.f, 0.f, 0.f, 0.f, 0.f, 0.f, 0.f};
#pragma unroll
      for (int kk = 0; kk < 4; ++kk) {
        int k0 = kk * 4 + 2 * hi;  // A 16x4: lanes 0-15 K={0,1}, 16-31 K={2,3}
        v2f a, bf;
        a.x = M[lo * 16 + k0];
        a.y = M[lo * 16 + k0 + 1];
        if constexpr (P == 0) {
          bf = *(const v2f*)(col + k0);  // contiguous pair -> b64 load
        } else {
          bf.x = col[k0 * STRIDE];
          bf.y = col[(k0 + 1) * STRIDE];
        }
        c = __builtin_amdgcn_wmma_f32_16x16x4_f32(false, a, false, bf,
                                                  (short)0, c, false, false);
      }
      if constexpr (P == 0) {  // 8 consecutive floats -> two b128 stores
        v4f s0 = {c[0], c[1], c[2], c[3]};
        v4f s1 = {c[4], c[5], c[6], c[7]};
        *(v4f*)(col + 8 * hi) = s0;
        *(v4f*)(col + 8 * hi + 4) = s1;
      } else {
#pragma unroll
        for (int v = 0; v < 8; ++v)  // D: VGPR v -> row v + 8*hi, col lo
          col[(v + 8 * hi) * STRIDE] = c[v];
      }
    } else {
      v8f c0 = {0.f, 0.f, 0.f, 0.f, 0.f, 0.f, 0.f, 0.f};
      v8f c1 = {0.f, 0.f, 0.f, 0.f, 0.f, 0.f, 0.f, 0.f};
#pragma unroll
      for (int kk = 0; kk < 8; ++kk) {
        int k0 = kk * 4 + 2 * hi;
        v2f bf, a0, a1;
        bf.x = col[k0 * STRIDE];
        bf.y = col[(k0 + 1) * STRIDE];
        a0.x = M[lo * 32 + k0];
        a0.y = M[lo * 32 + k0 + 1];
        a1.x = M[(16 + lo) * 32 + k0];
        a1.y = M[(16 + lo) * 32 + k0 + 1];
        c0 = __builtin_amdgcn_wmma_f32_16x16x4_f32(false, a0, false, bf,
                                                   (short)0, c0, false, false);
        c1 = __builtin_amdgcn_wmma_f32_16x16x4_f32(false, a1, false, bf,
                                                   (short)0, c1, false, false);
      }
#pragma unroll
      for (int v = 0; v < 8; ++v) {
        col[(v + 8 * hi) * STRIDE] = c0[v];
        col[(16 + v + 8 * hi) * STRIDE] = c1[v];
      }
    }
  }
}

// ---------------------------------------------------------------------------
// Measurement with folded final RY layer: <O_w> = cos(b2_w)<Z_w> - sin(b2_w)<X_w>
// low: contiguous 8K tiles -> Z(0..12), X(0..12), sum|amp|^2 (for high-Z signs)
// ---------------------------------------------------------------------------
__global__ void qe_measure_low(const float* __restrict__ state,
                               float* __restrict__ p1) {
  extern __shared__ float lds[];
  float* lre = lds;
  float* lim = lds + 8192;
  float* red = lds + 16384;
  int b = blockIdx.y, t = blockIdx.x;  // t = bits 13..17
  const float* re = state + ((long)(2 * b) << NQ) + (long)t * 8192;
  const float* im = state + ((long)(2 * b + 1) << NQ) + (long)t * 8192;
  for (int k = threadIdx.x; k < 8192; k += blockDim.x) {
    lre[k] = re[k];
    lim[k] = im[k];
  }
  __syncthreads();
  float acc[27];
#pragma unroll
  for (int q = 0; q < 27; ++q) acc[q] = 0.f;
  for (int k = threadIdx.x; k < 8192; k += blockDim.x) {
    float rv = lre[k], iv = lim[k];
    float P = rv * rv + iv * iv;
    acc[26] += P;
#pragma unroll
    for (int w = 0; w < 13; ++w) {
      acc[w] += ((k >> w) & 1) ? -P : P;
      int kp = k ^ (1 << w);
      acc[13 + w] += rv * lre[kp] + iv * lim[kp];  // sums to 2*Re<pair>
    }
  }
  for (int q = 0; q < 27; ++q) {
    __syncthreads();
    red[threadIdx.x] = acc[q];
    __syncthreads();
    for (int s = 128; s > 0; s >>= 1) {
      if ((int)threadIdx.x < s) red[threadIdx.x] += red[threadIdx.x + s];
      __syncthreads();
    }
    if (threadIdx.x == 0) p1[(b * 32 + t) * 28 + q] = red[0];
  }
}

// high: tiles covering bits {13..17} x {0..7} -> X(13..17)
__global__ void qe_measure_high(const float* __restrict__ state,
                                float* __restrict__ p2) {
  extern __shared__ float lds[];
  float* lre = lds;
  float* lim = lds + 8192;
  float* red = lds + 16384;
  int b = blockIdx.y, t = blockIdx.x;  // t = bits 8..12
  const float* re = state + ((long)(2 * b) << NQ);
  const float* im = state + ((long)(2 * b + 1) << NQ);
  for (int k = threadIdx.x; k < 8192; k += blockDim.x) {
    int idx = (k & 255) | (t << 8) | ((k >> 8) << 13);
    lre[k] = re[idx];
    lim[k] = im[idx];
  }
  __syncthreads();
  float acc[5];
#pragma unroll
  for (int q = 0; q < 5; ++q) acc[q] = 0.f;
  for (int k = threadIdx.x; k < 8192; k += blockDim.x) {
    float rv = lre[k], iv = lim[k];
#pragma unroll
    for (int q = 0; q < 5; ++q) {
      int kp = k ^ (256 << q);
      acc[q] += rv * lre[kp] + iv * lim[kp];
    }
  }
  for (int q = 0; q < 5; ++q) {
    __syncthreads();
    red[threadIdx.x] = acc[q];
    __syncthreads();
    for (int s = 128; s > 0; s >>= 1) {
      if ((int)threadIdx.x < s) red[threadIdx.x] += red[threadIdx.x + s];
      __syncthreads();
    }
    if (threadIdx.x == 0) p2[(b * 32 + t) * 8 + q] = red[0];
  }
}

__global__ void qe_reduce_out(const float* __restrict__ p1,
                              const float* __restrict__ p2,
                              const float* __restrict__ vp,
                              float* __restrict__ out) {
  int b = blockIdx.x, w = threadIdx.x;
  if (w >= NQ) return;
  float Z = 0.f, X = 0.f;
  if (w < 13) {
    for (int t = 0; t < 32; ++t) {
      const float* pp = p1 + (b * 32 + t) * 28;
      Z += pp[w];
      X += pp[13 + w];
    }
  } else {
    for (int t = 0; t < 32; ++t) {
      float Ps = p1[(b * 32 + t) * 28 + 26];
      Z += (((t >> (w - 13)) & 1) ? -Ps : Ps);
      X += p2[(b * 32 + t) * 8 + (w - 13)];
    }
  }
  float th = vp[(2 * NQ + w) * 2 + 1];        // b[2][w]
  out[b * NQ + w] = cosf(th) * Z - sinf(th) * X;  // RY^T Z RY = cZ - sX
}

extern "C" void kernel_launch(void* const* d_in, const int* in_sizes, int n_in,
                              void* d_out, int out_size, void* d_ws,
                              size_t ws_size, hipStream_t stream) {
  (void)in_sizes; (void)n_in; (void)out_size; (void)ws_size;
  const float* x  = (const float*)d_in[0];  // input_vec (64,32) f32
  const float* vp = (const float*)d_in[1];  // var_params (3,18,2) f32

  float* state = (float*)d_ws;                 // 128 * 2^18 f32 = 128 MB
  float* Mtab  = state + ((long)NCOMP << NQ);  // 3*3584 f32
  float* p1    = Mtab + 3 * 3584;              // 64*32*28 f32
  float* p2    = p1 + 64 * 32 * 28;            // 64*32*8 f32

  qe_setup_matrices<<<1, 256, 0, stream>>>(vp, Mtab);
  qe_init_state<<<dim3(64, 64), 256, 0, stream>>>(x, state);

  const int offs[5] = {0, 256, 1280, 2304, 3328};
  for (int r = 0; r < 3; ++r) {
    const float* Mr = Mtab + r * 3584;
    qe_stage<16, 0><<<2048, 256, 0, stream>>>(state, Mr + offs[0]);
    qe_stage<32, 3><<<2048, 256, 0, stream>>>(state, Mr + offs[1]);
    qe_stage<32, 7><<<2048, 256, 0, stream>>>(state, Mr + offs[2]);
    qe_stage<32, 11><<<2048, 256, 0, stream>>>(state, Mr + offs[3]);
    qe_stage<16, 14><<<2048, 256, 0, stream>>>(state, Mr + offs[4]);
  }

  size_t mshm = (size_t)(8192 * 2 + 256) * sizeof(float);  // 66560 B LDS
  qe_measure_low<<<dim3(32, 64), 256, mshm, stream>>>(state, p1);
  qe_measure_high<<<dim3(32, 64), 256, mshm, stream>>>(state, p2);
  qe_reduce_out<<<64, 32, 0, stream>>>(p1, p2, vp, (float*)d_out);
}